// GNNRegressor_81595788689871
// MI455X (gfx1250) — compile-verified
//
#include <hip/hip_runtime.h>
#include <hip/hip_bf16.h>
#include <stdint.h>

typedef __attribute__((ext_vector_type(2))) float v2f;
typedef __attribute__((ext_vector_type(8))) float v8f;

#define GNN_GRAPHS 64
#define GNN_OUTHID 64

// ---------------------------------------------------------------------------
// fp32 WMMA 16x16x4 wrapper (host pass never sees the amdgcn builtin)
// ---------------------------------------------------------------------------
__device__ __forceinline__ v8f wmma_f32_4(v2f a, v2f b, v8f c) {
#if defined(__HIP_DEVICE_COMPILE__)
  return __builtin_amdgcn_wmma_f32_16x16x4_f32(
      /*neg_a=*/false, a, /*neg_b=*/false, b,
      /*c_mod=*/(short)0, c, /*reuse_a=*/false, /*reuse_b=*/false);
#else
  (void)a; (void)b;
  return c;
#endif
}

// ---------------------------------------------------------------------------
// Degree / normalization
// ---------------------------------------------------------------------------
__global__ void k_init_deg(float* deg, int N) {
  int i = blockIdx.x * blockDim.x + threadIdx.x;
  if (i < N) deg[i] = 1.0f;  // self-loop contributes 1
}

__global__ void k_accum_deg(const int* __restrict__ ei, float* deg, long long E) {
  long long e = blockIdx.x * (long long)blockDim.x + threadIdx.x;
  if (e < E) atomicAdd(&deg[ei[E + e]], 1.0f);  // dst row of edge_index
}

__global__ void k_rsqrt_deg(float* dinv, int N) {
  int i = blockIdx.x * blockDim.x + threadIdx.x;
  if (i < N) dinv[i] = rsqrtf(fmaxf(dinv[i], 1.0f));
}

// ---------------------------------------------------------------------------
// Layer-1 GEMM: t = x @ W1   (K = 5, too small for WMMA; scalar FMA)
// ---------------------------------------------------------------------------
__global__ void k_gemm_small(const float* __restrict__ X, const float* __restrict__ W,
                             float* __restrict__ H, int N, int K, int logF) {
  int F = 1 << logF;
  long long idx = blockIdx.x * (long long)blockDim.x + threadIdx.x;
  if (idx >= (long long)N << logF) return;
  int i = (int)(idx >> logF);
  int f = (int)(idx & (F - 1));
  float acc = 0.0f;
  for (int k = 0; k < K; ++k) acc = fmaf(X[(size_t)i * K + k], W[(size_t)k << logF | f], acc);
  H[idx] = acc;
}

// ---------------------------------------------------------------------------
// WMMA GEMM: H[N,F] = X[N,128] @ W[128,F]
//  - W staged once per block into LDS, PAIR-INTERLEAVED:
//        sW[(kp*F + n)*2 + p] = W[2*kp + p][n]
//    so each B fragment {row k, row k+1} at column n is ONE aligned
//    ds_load_b64 straight into an even VGPR pair (no reassembly movs).
//  - A fragment is one global_load_b64 (addresses 8B-aligned: K=128, koff even).
//  - K-loop unroll capped at 2 -> ~110 VGPRs, no scratch spills.
//  - Tile guard after __syncthreads and wave-uniform: EXEC all-1s at WMMAs.
// ---------------------------------------------------------------------------
template <int NCT>
__global__ void k_gemm_wmma(const float* __restrict__ X, const float* __restrict__ W,
                            float* __restrict__ H, int nrows) {
  constexpr int K = 128;
  constexpr int F = NCT * 16;
  constexpr int PAIRS = K / 2;
  __shared__ float sW[K * F];

  // cooperative stage + interleave: thread handles (kp, n..n+3)
  for (int i = threadIdx.x * 4; i < PAIRS * F; i += blockDim.x * 4) {
    int kp = i / F;      // K-pair index (F is a power of two -> shift)
    int n  = i & (F - 1);
    float4 r0 = *(const float4*)&W[(size_t)(2 * kp) * F + n];
    float4 r1 = *(const float4*)&W[(size_t)(2 * kp + 1) * F + n];
    float4 lo = make_float4(r0.x, r1.x, r0.y, r1.y);
    float4 hi = make_float4(r0.z, r1.z, r0.w, r1.w);
    *(float4*)&sW[(size_t)i * 2]     = lo;
    *(float4*)&sW[(size_t)i * 2 + 4] = hi;
  }
  __syncthreads();

  int wave = threadIdx.x >> 5;
  int lane = threadIdx.x & 31;
  int ntiles = nrows >> 4;  // nrows is a multiple of 16 (50000 = 3125*16)
  int tile = blockIdx.x * (blockDim.x >> 5) + wave;
  if (tile >= ntiles) return;  // whole wave exits together

  int row0 = tile << 4;
  int m    = lane & 15;        // M for A frag / N for B,D frags
  int half = lane >> 4;        // 0: K={0,1}   1: K={2,3}
  int koff = half << 1;

  v8f acc[NCT];
#pragma unroll
  for (int c = 0; c < NCT; ++c) acc[c] = (v8f){};

  const float* xrow = X + (size_t)(row0 + m) * K;
#pragma unroll 2
  for (int k = 0; k < K; k += 4) {
    v2f a = *(const v2f*)&xrow[k + koff];
    const float* wp = &sW[(size_t)(((k + koff) >> 1) * F + m) * 2];
#pragma unroll
    for (int c = 0; c < NCT; ++c) {
      v2f b = *(const v2f*)&wp[c << 5];  // 16 columns * 2 floats per col tile
      acc[c] = wmma_f32_4(a, b, acc[c]);
    }
  }

  int mbase = half << 3;  // D: lanes 16-31 hold M = vgpr + 8
#pragma unroll
  for (int c = 0; c < NCT; ++c) {
    float* hp = H + (size_t)row0 * F + (c << 4) + m;
#pragma unroll
    for (int v = 0; v < 8; ++v) hp[(size_t)(mbase + v) * F] = acc[c][v];
  }
}

// ---------------------------------------------------------------------------
// Aggregation: agg = selfloop + scatter_add over edges, then bias+ReLU
// ---------------------------------------------------------------------------
__global__ void k_selfloop(const float* __restrict__ t, const float* __restrict__ dinv,
                           float* __restrict__ agg, int N, int logF) {
  long long idx = blockIdx.x * (long long)blockDim.x + threadIdx.x;
  if (idx >= (long long)N << logF) return;
  int i = (int)(idx >> logF);
  float d = dinv[i];
  agg[idx] = t[idx] * d * d;
}

__global__ void k_edge_agg(const int* __restrict__ ei, const float* __restrict__ t,
                           const float* __restrict__ dinv, float* __restrict__ agg,
                           long long E, int logF) {
  int F = 1 << logF;
  long long idx = blockIdx.x * (long long)blockDim.x + threadIdx.x;
  if (idx >= E << logF) return;
  long long e = idx >> logF;
  int f = (int)(idx & (F - 1));
  int s = ei[e];
  int d = ei[E + e];
  float w = dinv[s] * dinv[d];
  atomicAdd(&agg[((size_t)d << logF) | f], t[((size_t)s << logF) | f] * w);
}

__global__ void k_bias_relu(float* __restrict__ h, const float* __restrict__ b,
                            int N, int logF) {
  int F = 1 << logF;
  long long idx = blockIdx.x * (long long)blockDim.x + threadIdx.x;
  if (idx >= (long long)N << logF) return;
  int f = (int)(idx & (F - 1));
  h[idx] = fmaxf(h[idx] + b[f], 0.0f);
}

// ---------------------------------------------------------------------------
// Global mean pool + final FC
// ---------------------------------------------------------------------------
__global__ void k_zero(float* p, int n) {
  int i = blockIdx.x * blockDim.x + threadIdx.x;
  if (i < n) p[i] = 0.0f;
}

__global__ void k_pool_sum(const int* __restrict__ batch, const float* __restrict__ h,
                           float* __restrict__ pool, int N) {
  long long idx = blockIdx.x * (long long)blockDim.x + threadIdx.x;
  if (idx >= (long long)N * GNN_OUTHID) return;
  int i = (int)(idx >> 6);
  int f = (int)(idx & 63);
  atomicAdd(&pool[(size_t)batch[i] * GNN_OUTHID + f], h[idx]);
}

__global__ void k_pool_cnt(const int* __restrict__ batch, float* __restrict__ cnt, int N) {
  int i = blockIdx.x * blockDim.x + threadIdx.x;
  if (i < N) atomicAdd(&cnt[batch[i]], 1.0f);
}

__global__ void k_final(const float* __restrict__ pool, const float* __restrict__ cnt,
                        const float* __restrict__ Wfc, const float* __restrict__ bfc,
                        float* __restrict__ out) {
  int g = threadIdx.x;
  if (g >= GNN_GRAPHS) return;
  float acc = 0.0f;
#pragma unroll
  for (int f = 0; f < GNN_OUTHID; ++f) acc = fmaf(pool[g * GNN_OUTHID + f], Wfc[f], acc);
  out[g] = acc / fmaxf(cnt[g], 1.0f) + bfc[0];
}

// ---------------------------------------------------------------------------
// Host launch
// ---------------------------------------------------------------------------
static inline unsigned gdim(long long work, int block) {
  return (unsigned)((work + block - 1) / block);
}

extern "C" void kernel_launch(void* const* d_in, const int* in_sizes, int n_in,
                              void* d_out, int out_size, void* d_ws, size_t ws_size,
                              hipStream_t stream) {
  const float* x    = (const float*)d_in[0];
  const int*   ei   = (const int*)d_in[1];
  const int*   batc = (const int*)d_in[2];
  const float* W1   = (const float*)d_in[3];
  const float* b1   = (const float*)d_in[4];
  const float* W2   = (const float*)d_in[5];
  const float* b2   = (const float*)d_in[6];
  const float* W3   = (const float*)d_in[7];
  const float* b3   = (const float*)d_in[8];
  const float* Wfc  = (const float*)d_in[9];
  const float* bfc  = (const float*)d_in[10];
  float* out = (float*)d_out;

  const int       N = in_sizes[0] / 5;        // 50000
  const long long E = in_sizes[1] / 2;        // 800000
  const int HID = 128, LOG_HID = 7, LOG_OUT = 6;

  // workspace layout (floats): dinv | bufA (N*128) | bufB (N*128) | pool | cnt
  float* ws   = (float*)d_ws;
  float* dinv = ws;
  float* bufA = ws + (size_t)((N + 63) / 64) * 64;
  float* bufB = bufA + (size_t)N * HID;
  float* pool = bufB + (size_t)N * HID;
  float* cnt  = pool + GNN_GRAPHS * GNN_OUTHID;

  const int BLK = 256;

  // 1) symmetric normalization dinv = rsqrt(deg), deg includes self-loop
  k_init_deg<<<gdim(N, BLK), BLK, 0, stream>>>(dinv, N);
  k_accum_deg<<<gdim(E, BLK), BLK, 0, stream>>>(ei, dinv, E);
  k_rsqrt_deg<<<gdim(N, BLK), BLK, 0, stream>>>(dinv, N);

  // 2) layer 1: t = x @ W1 (K=5); aggregate; bias+relu   -> bufB [N,128]
  k_gemm_small<<<gdim((long long)N << LOG_HID, BLK), BLK, 0, stream>>>(x, W1, bufA, N, 5, LOG_HID);
  k_selfloop<<<gdim((long long)N << LOG_HID, BLK), BLK, 0, stream>>>(bufA, dinv, bufB, N, LOG_HID);
  k_edge_agg<<<gdim(E << LOG_HID, BLK), BLK, 0, stream>>>(ei, bufA, dinv, bufB, E, LOG_HID);
  k_bias_relu<<<gdim((long long)N << LOG_HID, BLK), BLK, 0, stream>>>(bufB, b1, N, LOG_HID);

  // 3) layer 2: t = h1 @ W2 via fp32 WMMA (F=128 -> 8 col tiles, W2 in LDS)
  k_gemm_wmma<8><<<gdim((N >> 4), 8), 8 * 32, 0, stream>>>(bufB, W2, bufA, N);
  k_selfloop<<<gdim((long long)N << LOG_HID, BLK), BLK, 0, stream>>>(bufA, dinv, bufB, N, LOG_HID);
  k_edge_agg<<<gdim(E << LOG_HID, BLK), BLK, 0, stream>>>(ei, bufA, dinv, bufB, E, LOG_HID);
  k_bias_relu<<<gdim((long long)N << LOG_HID, BLK), BLK, 0, stream>>>(bufB, b2, N, LOG_HID);

  // 4) layer 3: t = h2 @ W3 via fp32 WMMA (F=64 -> 4 col tiles, W3 in LDS)
  k_gemm_wmma<4><<<gdim((N >> 4), 8), 8 * 32, 0, stream>>>(bufB, W3, bufA, N);
  k_selfloop<<<gdim((long long)N << LOG_OUT, BLK), BLK, 0, stream>>>(bufA, dinv, bufB, N, LOG_OUT);
  k_edge_agg<<<gdim(E << LOG_OUT, BLK), BLK, 0, stream>>>(ei, bufA, dinv, bufB, E, LOG_OUT);
  k_bias_relu<<<gdim((long long)N << LOG_OUT, BLK), BLK, 0, stream>>>(bufB, b3, N, LOG_OUT);

  // 5) global mean pool + FC
  k_zero<<<gdim(GNN_GRAPHS * GNN_OUTHID + GNN_GRAPHS, BLK), BLK, 0, stream>>>(
      pool, GNN_GRAPHS * GNN_OUTHID + GNN_GRAPHS);
  k_pool_sum<<<gdim((long long)N * GNN_OUTHID, BLK), BLK, 0, stream>>>(batc, bufB, pool, N);
  k_pool_cnt<<<gdim(N, BLK), BLK, 0, stream>>>(batc, cnt, N);
  k_final<<<1, GNN_GRAPHS, 0, stream>>>(pool, cnt, Wfc, bfc, out);
}